// GroverEncoderWrapper_24824910971087
// MI455X (gfx1250) — compile-verified
//
#include <hip/hip_runtime.h>

#define NA 50000
#define NBOND 100000
#define H 256
#define MAXNB 6
#define DEPTH 6

#define BM 64
#define KC 32
#define LDA 36   // padded row stride (floats) -> conflict-free ds_load_b64 frags
#define LDB 36

typedef __attribute__((ext_vector_type(2))) float v2f;
typedef __attribute__((ext_vector_type(8))) float v8f;

// ---------------- per-atom neighbor aggregation: nei[a] = sum_j msg[a2b[a][j]]
__global__ __launch_bounds__(256) void agg_kernel(const float* __restrict__ msg,
                                                  const int* __restrict__ a2b,
                                                  float* __restrict__ nei) {
    int idx  = blockIdx.x * 256 + threadIdx.x;
    int atom = idx >> 6;            // 64 float4 chunks per row
    int c    = (idx & 63) << 2;
    if (atom >= NA) return;
    const int* ab = a2b + atom * MAXNB;
    float4 acc = make_float4(0.f, 0.f, 0.f, 0.f);
#pragma unroll
    for (int j = 0; j < MAXNB; ++j) {
        int b = ab[j];
        float4 v = *(const float4*)(msg + (size_t)b * H + c);
        acc.x += v.x; acc.y += v.y; acc.z += v.z; acc.w += v.w;
    }
    *(float4*)(nei + (size_t)atom * H + c) = acc;
}

// ---------------- one-time weight transpose: Wt[n][k] = W[k][n]
__global__ __launch_bounds__(256) void transpose_kernel(const float* __restrict__ W,
                                                        float* __restrict__ Wt) {
    int idx = blockIdx.x * 256 + threadIdx.x;   // 65536 total
    int k = idx >> 8;
    int n = idx & 255;
    Wt[(size_t)n * H + k] = W[(size_t)k * H + n];
}

// ---------------- fused: out[b] = relu(inp[b] + (nei[b2a[b]] - msg[b2revb[b]]) @ W)
// Software-pipelined: global loads for chunk c+1 are issued before the WMMA
// compute of chunk c so VMEM latency hides under the matrix pipe.
__global__ __launch_bounds__(256) void mpn_gemm_kernel(
    const float* __restrict__ nei,    // [NA][H]
    const float* __restrict__ msg,    // [NBOND][H]
    const float* __restrict__ inp,    // [NBOND][H] (f_bonds, skip input)
    const float* __restrict__ Wt,     // [H][H] transposed: Wt[n][k]
    const int*   __restrict__ b2a,
    const int*   __restrict__ b2revb,
    float*       __restrict__ out)    // [NBOND][H]
{
    __shared__ __align__(16) float As[BM * LDA];   // 64 x 32 (+pad)
    __shared__ __align__(16) float Bs[H * LDB];    // 256 x 32 (+pad)
    __shared__ int sA[BM];
    __shared__ int sR[BM];

    const int tid  = threadIdx.x;
    const int base = blockIdx.x * BM;

    if (tid < BM) {
        int b = base + tid;
        if (b >= NBOND) b = NBOND - 1;
        sA[tid] = b2a[b];
        sR[tid] = b2revb[b];
    }
    __syncthreads();

    // Per-thread staging addresses (k0 added per chunk).
    const float* gA[2]; const float* gR[2]; float* lA[2];
#pragma unroll
    for (int i = 0; i < 2; ++i) {
        int fid = tid + i * 256;          // 64 rows x 8 float4
        int row = fid >> 3;
        int kq  = (fid & 7) << 2;
        gA[i] = nei + (size_t)sA[row] * H + kq;
        gR[i] = msg + (size_t)sR[row] * H + kq;
        lA[i] = As + row * LDA + kq;
    }
    const float* gB[8]; float* lB[8];
#pragma unroll
    for (int i = 0; i < 8; ++i) {
        int fid = tid + i * 256;          // 256 n x 8 float4
        int n  = fid >> 3;
        int kq = (fid & 7) << 2;
        gB[i] = Wt + (size_t)n * H + kq;
        lB[i] = Bs + n * LDB + kq;
    }

    const int lane = tid & 31;
    const int wave = tid >> 5;
    const int mt   = wave >> 1;       // 0..3 : 16-row M tile
    const int nh   = wave & 1;        // 0..1 : 128-col N half
    const int half = lane >> 4;       // K sub-pair select (per ISA 16x4 layout)
    const int l16  = lane & 15;
    const float* fragA = As + (mt * 16 + l16) * LDA + half * 2;
    const float* fragB = Bs + (nh * 128 + l16) * LDB + half * 2;

    v8f acc[8] = {};                  // 8 x (16x16 f32) accumulator tiles
    float4 rNa[2], rNr[2], rB[8];     // register staging for next chunk

    // prologue: issue chunk-0 global loads
#pragma unroll
    for (int i = 0; i < 2; ++i) {
        rNa[i] = *(const float4*)(gA[i]);
        rNr[i] = *(const float4*)(gR[i]);
    }
#pragma unroll
    for (int i = 0; i < 8; ++i) rB[i] = *(const float4*)(gB[i]);

    const int NCHUNK = H / KC;
    for (int c = 0; c < NCHUNK; ++c) {
        // drain staged registers into LDS (single batched wait on loads)
#pragma unroll
        for (int i = 0; i < 2; ++i) {
            float4 d;
            d.x = rNa[i].x - rNr[i].x; d.y = rNa[i].y - rNr[i].y;
            d.z = rNa[i].z - rNr[i].z; d.w = rNa[i].w - rNr[i].w;
            *(float4*)(lA[i]) = d;
        }
#pragma unroll
        for (int i = 0; i < 8; ++i) *(float4*)(lB[i]) = rB[i];
        __syncthreads();

        // issue next chunk's global loads; they fly while WMMAs run below
        if (c + 1 < NCHUNK) {
            const int k0 = (c + 1) * KC;
#pragma unroll
            for (int i = 0; i < 2; ++i) {
                rNa[i] = *(const float4*)(gA[i] + k0);
                rNr[i] = *(const float4*)(gR[i] + k0);
            }
#pragma unroll
            for (int i = 0; i < 8; ++i) rB[i] = *(const float4*)(gB[i] + k0);
        }

#pragma unroll
        for (int kc = 0; kc < KC; kc += 4) {
            // A frag (16x4 f32): lanes 0-15 K={kc,kc+1}, lanes 16-31 K={kc+2,kc+3}
            float2 af = *(const float2*)(fragA + kc);
            v2f a; a.x = af.x; a.y = af.y;
#pragma unroll
            for (int j = 0; j < 8; ++j) {
                float2 bf = *(const float2*)(fragB + j * 16 * LDB + kc);
                v2f b; b.x = bf.x; b.y = bf.y;
                acc[j] = __builtin_amdgcn_wmma_f32_16x16x4_f32(
                    false, a, false, b, (short)0, acc[j], false, false);
            }
        }
        __syncthreads();
    }

    // epilogue: skip + relu + store (C layout: VGPR r -> M = half*8 + r, N = l16)
#pragma unroll
    for (int j = 0; j < 8; ++j) {
        int col = (nh * 8 + j) * 16 + l16;
#pragma unroll
        for (int r = 0; r < 8; ++r) {
            int bond = base + mt * 16 + half * 8 + r;
            if (bond < NBOND) {
                size_t off = (size_t)bond * H + col;
                float v = acc[j][r] + inp[off];
                out[off] = v > 0.f ? v : 0.f;
            }
        }
    }
}

extern "C" void kernel_launch(void* const* d_in, const int* in_sizes, int n_in,
                              void* d_out, int out_size, void* d_ws, size_t ws_size,
                              hipStream_t stream) {
    const float* f_bonds = (const float*)d_in[0];
    const int*   a2b     = (const int*)d_in[1];
    const int*   b2a     = (const int*)d_in[2];
    const int*   b2revb  = (const int*)d_in[3];
    const float* Wm[3]   = {(const float*)d_in[4],
                            (const float*)d_in[5],
                            (const float*)d_in[6]};

    float* out    = (float*)d_out;
    float* nei    = (float*)d_ws;                       // NA*H
    float* msgbuf = nei + (size_t)NA * H;               // NBOND*H
    float* Wt     = msgbuf + (size_t)NBOND * H;         // H*H

    dim3 blk(256);
    dim3 aggGrid((NA * (H / 4) + 255) / 256);
    dim3 trGrid((H * H) / 256);
    dim3 gemmGrid((NBOND + BM - 1) / BM);

    for (int r = 0; r < 3; ++r) {
        float* outslice = out + (size_t)r * NBOND * H;
        transpose_kernel<<<trGrid, blk, 0, stream>>>(Wm[r], Wt);
        const float* src = f_bonds;
        for (int it = 0; it < DEPTH - 1; ++it) {
            float* dst = (it & 1) ? msgbuf : outslice;  // final (it=4) -> outslice
            agg_kernel<<<aggGrid, blk, 0, stream>>>(src, a2b, nei);
            mpn_gemm_kernel<<<gemmGrid, blk, 0, stream>>>(
                nei, src, f_bonds, Wt, b2a, b2revb, dst);
            src = dst;
        }
    }
}